// WaveletNet_90108413870128
// MI455X (gfx1250) — compile-verified
//
#include <hip/hip_runtime.h>

typedef unsigned int       u32;
typedef unsigned long long u64;
typedef __attribute__((ext_vector_type(4))) u32   u32x4;
typedef __attribute__((ext_vector_type(8))) int   i32x8;
typedef __attribute__((ext_vector_type(4))) int   i32x4;
typedef __attribute__((ext_vector_type(4))) float f32x4;

#define TILE_W 128   // input tile width  (floats)
#define TILE_H 32    // input tile height (rows)

// ---------------------------------------------------------------------------
// TDM: DMA a (tw x th) float tile (row pitch w elements) from global into LDS.
// Must be executed by a single wave; caller barriers afterwards.
// ---------------------------------------------------------------------------
template <int CPOL>
__device__ __forceinline__ void tdm_load_tile(const float* gsrc, u32 lds_off,
                                              int w, int h, int tw, int th)
{
    u64 ga = (u64)(size_t)gsrc;

    // D# group 0: count=1 | lds_addr | global_addr[56:0] | type=2
    u32x4 g0;
    g0[0] = 1u;
    g0[1] = lds_off;
    g0[2] = (u32)ga;
    g0[3] = (u32)((ga >> 32) & 0x1FFFFFFu) | (2u << 30);

    // D# group 1: data_size=4B; tensor_dim0=w, tensor_dim1=h;
    // tile_dim0=tw, tile_dim1=th (tile_dim2=0); tensor_dim0_stride=w.
    i32x8 g1;
    g1[0] = (int)(2u << 16);
    g1[1] = (int)(((u32)w & 0xFFFFu) << 16);
    g1[2] = (int)(((u32)w >> 16) | (((u32)h & 0xFFFFu) << 16));
    g1[3] = (int)(((u32)h >> 16) | ((u32)tw << 16));
    g1[4] = (int)(u32)th;
    g1[5] = (int)(u32)w;
    g1[6] = 0;
    g1[7] = 0;

    i32x4 gz = {0, 0, 0, 0};               // groups 2/3: 2-D tensor, unused dims
#if defined(__clang_major__) && (__clang_major__ >= 23)
    i32x8 gz8 = {0, 0, 0, 0, 0, 0, 0, 0};
    __builtin_amdgcn_tensor_load_to_lds(g0, g1, gz, gz, gz8, CPOL);
#else
    __builtin_amdgcn_tensor_load_to_lds(g0, g1, gz, gz, CPOL);
#endif
    __builtin_amdgcn_s_wait_tensorcnt(0);
}

// 2x2 Haar on 4 horizontally adjacent blocks held in two row-pair f32x4 pairs.
__device__ __forceinline__ void haar4(const f32x4 p0, const f32x4 p1,
                                      const f32x4 q0, const f32x4 q1,
                                      f32x4& ll, f32x4& lh, f32x4& hl, f32x4& hh)
{
    const float av[4] = {p0.x, p0.z, p1.x, p1.z};
    const float bv[4] = {p0.y, p0.w, p1.y, p1.w};
    const float cv[4] = {q0.x, q0.z, q1.x, q1.z};
    const float dv[4] = {q0.y, q0.w, q1.y, q1.w};
#pragma unroll
    for (int k = 0; k < 4; ++k) {
        const float a = av[k], b = bv[k], c = cv[k], d = dv[k];
        const float u = __builtin_fabsf(a - c);
        const float v = __builtin_fabsf(b - d);
        ll[k] = (a + b + c + d) * 0.25f;
        lh[k] = __builtin_fabsf((a + c) - (b + d)) * 0.5f;
        hl[k] = (u + v) * 0.5f;
        hh[k] = __builtin_fabsf(u - v);
    }
}

// ---------------------------------------------------------------------------
// One Haar level (levels 0 and 1): tiled TDM -> LDS -> subband stores.
// CPOL: TDM load cache policy (1 = TH_NT for the read-once level-0 input).
// ---------------------------------------------------------------------------
template <int CPOL>
__global__ __launch_bounds__(256) void haar_level_kernel(
    const float* __restrict__ in_base, long in_plane_stride,
    int w, int h, float* __restrict__ out_base)
{
    __shared__ float tile[TILE_H * TILE_W];   // 16 KB

    const int bc  = blockIdx.z;
    const int tx0 = blockIdx.x * TILE_W;
    const int ty0 = blockIdx.y * TILE_H;
    const int tw  = min(TILE_W, w - tx0);     // power of two for all levels
    const int th  = min(TILE_H, h - ty0);

    const float* gsrc = in_base + (long)bc * in_plane_stride + (long)ty0 * w + tx0;

    if (threadIdx.x < 32u) {   // wave 0 only
        tdm_load_tile<CPOL>(gsrc, (u32)(size_t)(void*)tile, w, h, tw, th);
    }
    __syncthreads();

    const int ow  = tw >> 1;
    const int oh  = th >> 1;
    const int lxq = (int)(threadIdx.x & 15u);
    const int ly  = (int)(threadIdx.x >> 4);
    const int lx  = lxq << 2;

    if (lx < ow && ly < oh) {
        const float* r0 = &tile[(2 * ly) * tw + 2 * lx];
        const float* r1 = r0 + tw;
        f32x4 p0 = *(const f32x4*)(r0);
        f32x4 p1 = *(const f32x4*)(r0 + 4);
        f32x4 q0 = *(const f32x4*)(r1);
        f32x4 q1 = *(const f32x4*)(r1 + 4);

        f32x4 ll, lh, hl, hh;
        haar4(p0, p1, q0, q1, ll, lh, hl, hh);

        const int  w2    = w >> 1;
        const long plane = (long)(h >> 1) * (long)w2;
        const int  gx    = blockIdx.x * (TILE_W / 2) + lx;
        const int  gy    = blockIdx.y * (TILE_H / 2) + ly;

        float* o = out_base + (long)bc * 4 * plane + (long)gy * w2 + gx;
        *(f32x4*)o = ll;                                          // LL re-read by next level
        __builtin_nontemporal_store(lh, (f32x4*)(o +     plane)); // detail bands final: th:NT
        __builtin_nontemporal_store(hl, (f32x4*)(o + 2 * plane));
        __builtin_nontemporal_store(hh, (f32x4*)(o + 3 * plane));
    }
}

// ---------------------------------------------------------------------------
// Fused tail: levels 2,3,4 fully LDS-resident. One workgroup per (bc) plane.
// Input: 128x128 LL planes of level 1. LL2/LL3 chain through LDS (also stored to out2/out3).
// ---------------------------------------------------------------------------
__device__ __forceinline__ void haar_step_lds(
    const float* src, int sw, int sh,   // LDS source, pitch sw, height sh
    float* lds_ll,                      // LDS dst for LL (pitch sw/2), or nullptr
    float* gout, long plane)            // global plane base (subband 0), plane size
{
    const int ow    = sw >> 1;
    const int oh    = sh >> 1;
    const int qpr   = ow >> 2;          // f32x4 quads per output row
    const int total = oh * qpr;
    for (int i = (int)threadIdx.x; i < total; i += 256) {
        const int qy = i / qpr;
        const int lx = (i - qy * qpr) << 2;

        const float* r0 = src + (2 * qy) * sw + 2 * lx;
        const float* r1 = r0 + sw;
        f32x4 p0 = *(const f32x4*)(r0);
        f32x4 p1 = *(const f32x4*)(r0 + 4);
        f32x4 q0 = *(const f32x4*)(r1);
        f32x4 q1 = *(const f32x4*)(r1 + 4);

        f32x4 ll, lh, hl, hh;
        haar4(p0, p1, q0, q1, ll, lh, hl, hh);

        float* o = gout + (long)qy * ow + lx;
        __builtin_nontemporal_store(ll, (f32x4*)(o));             // all tail outputs final
        __builtin_nontemporal_store(lh, (f32x4*)(o +     plane));
        __builtin_nontemporal_store(hl, (f32x4*)(o + 2 * plane));
        __builtin_nontemporal_store(hh, (f32x4*)(o + 3 * plane));
        if (lds_ll) *(f32x4*)(lds_ll + qy * ow + lx) = ll;        // keep LL chain in LDS
    }
}

__global__ __launch_bounds__(256) void haar_fused_tail_kernel(
    const float* __restrict__ in_base, long in_plane_stride,
    float* __restrict__ out2, float* __restrict__ out3, float* __restrict__ out4)
{
    __shared__ float t0[128 * 128];   // 64 KB : level-2 input
    __shared__ float t1[64 * 64];     // 16 KB : LL2
    __shared__ float t2[32 * 32];     //  4 KB : LL3

    const int bc = blockIdx.x;
    const float* gsrc = in_base + (long)bc * in_plane_stride;

    if (threadIdx.x < 32u) {   // wave 0: DMA whole 128x128 plane (RT: producer just wrote it)
        tdm_load_tile<0>(gsrc, (u32)(size_t)(void*)t0, 128, 128, 128, 128);
    }
    __syncthreads();

    haar_step_lds(t0, 128, 128, t1, out2 + (long)bc * 4 * 4096, 4096);  // level 2 -> 64x64
    __syncthreads();
    haar_step_lds(t1, 64, 64, t2, out3 + (long)bc * 4 * 1024, 1024);    // level 3 -> 32x32
    __syncthreads();
    haar_step_lds(t2, 32, 32, nullptr, out4 + (long)bc * 4 * 256, 256); // level 4 -> 16x16
}

// ---------------------------------------------------------------------------
extern "C" void kernel_launch(void* const* d_in, const int* in_sizes, int n_in,
                              void* d_out, int out_size, void* d_ws, size_t ws_size,
                              hipStream_t stream) {
    (void)in_sizes; (void)n_in; (void)out_size; (void)d_ws; (void)ws_size;

    const float* in0 = (const float*)d_in[0];
    float*       out = (float*)d_out;

    const long S0 = 384L * 65536;   // level-0 output elements (96 planes * 4 * 256*256)
    const long S1 = 384L * 16384;
    const long S2 = 384L * 4096;
    const long S3 = 384L * 1024;

    // Level 0: 512x512 -> 256x256, input read once -> NT load policy.
    haar_level_kernel<1><<<dim3(4, 16, 96), 256, 0, stream>>>(
        in0, 512L * 512, 512, 512, out);

    // Level 1: reads LL of level 0 (hot in L2) -> RT load policy.
    haar_level_kernel<0><<<dim3(2, 8, 96), 256, 0, stream>>>(
        out, 4L * 256 * 256, 256, 256, out + S0);

    // Levels 2-4 fused, LDS-resident (one block per plane).
    haar_fused_tail_kernel<<<dim3(96), 256, 0, stream>>>(
        out + S0, 4L * 128 * 128,
        out + S0 + S1,
        out + S0 + S1 + S2,
        out + S0 + S1 + S2 + S3);
}